// SimpleGATLayer_1992864825875
// MI455X (gfx1250) — compile-verified
//
#include <hip/hip_runtime.h>

typedef __attribute__((ext_vector_type(2))) float v2f;
typedef __attribute__((ext_vector_type(8))) float v8f;
typedef __attribute__((ext_vector_type(4))) int   v4i;

// address-space-qualified pointee types for the async-DMA builtin
typedef __attribute__((address_space(1))) v4i gv4i;   // global
typedef __attribute__((address_space(3))) v4i lv4i;   // LDS

#define B_    8
#define N_    2048
#define FIN_  256
#define FOUT_ 128
#define NT_   (N_ / 16)   // 128 j-tiles per batch row-block
#define TSTR  132         // padded LDS row stride (dwords): conflict-free dual reads

__device__ __forceinline__ float lrelu(float x) {
    return x > 0.f ? x : 0.01f * x;
}

// ---- gfx1250 async global->LDS copy (ASYNCcnt) with sync fallback ----------
__device__ __forceinline__ void async_b128(const float* g, float* l) {
#if __has_builtin(__builtin_amdgcn_global_load_async_to_lds_b128)
    __builtin_amdgcn_global_load_async_to_lds_b128(
        (gv4i*)g, (lv4i*)l, 0, 0);
#else
    *(float4*)l = *(const float4*)g;
#endif
}

template <int N>
__device__ __forceinline__ void wait_async() {
#if __has_builtin(__builtin_amdgcn_s_wait_asynccnt)
    __builtin_amdgcn_s_wait_asynccnt(N);
#else
    if (N == 0) asm volatile("s_wait_asynccnt 0x0" ::: "memory");
    else        asm volatile("s_wait_asynccnt 0x4" ::: "memory");
#endif
}

// ---------------------------------------------------------------------------
// Kernel 1: Wh = h @ W via v_wmma_f32_16x16x4_f32.
// grid = B*N/16 row-tiles, 256 threads (8 waves); wave w owns col-tile w.
// ---------------------------------------------------------------------------
__global__ __launch_bounds__(256) void wh_kernel(const float* __restrict__ h,
                                                 const float* __restrict__ W,
                                                 float* __restrict__ Wh) {
    const int row0 = blockIdx.x * 16;
    const int wave = threadIdx.x >> 5;
    const int lane = threadIdx.x & 31;
    const int half = lane >> 4;
    const int lm   = lane & 15;
    const int n0   = wave * 16;

    v8f acc = {};
    const float* arow = h + (size_t)(row0 + lm) * FIN_;
    for (int k = 0; k < FIN_; k += 4) {
        v2f a = *(const v2f*)(arow + k + 2 * half);
        v2f b;
        const float* bp = W + (size_t)(k + 2 * half) * FOUT_ + n0 + lm;
        b.x = bp[0];
        b.y = bp[FOUT_];
        acc = __builtin_amdgcn_wmma_f32_16x16x4_f32(false, a, false, b,
                                                    (short)0, acc, false, false);
    }
    float* orow = Wh + (size_t)(row0 + 8 * half) * FOUT_ + n0 + lm;
#pragma unroll
    for (int r = 0; r < 8; ++r)
        orow[(size_t)r * FOUT_] = acc[r];
}

// ---------------------------------------------------------------------------
// Kernel 2a: s_i / s_j row dots; one wave per row, shfl-xor reduce (wave32).
// ---------------------------------------------------------------------------
__global__ __launch_bounds__(256) void score_kernel(const float* __restrict__ Wh,
                                                    const float* __restrict__ a_i,
                                                    const float* __restrict__ a_j,
                                                    float* __restrict__ s_i,
                                                    float* __restrict__ s_j) {
    const int wave = threadIdx.x >> 5;
    const int lane = threadIdx.x & 31;
    const int row  = blockIdx.x * 8 + wave;
    const float* wr = Wh + (size_t)row * FOUT_;
    float si = 0.f, sj = 0.f;
#pragma unroll
    for (int c = 0; c < 4; ++c) {
        float v = wr[lane + 32 * c];
        si += v * a_i[lane + 32 * c];
        sj += v * a_j[lane + 32 * c];
    }
#pragma unroll
    for (int off = 16; off > 0; off >>= 1) {
        si += __shfl_xor(si, off, 32);
        sj += __shfl_xor(sj, off, 32);
    }
    if (lane == 0) { s_i[row] = si; s_j[row] = sj; }
}

// ---------------------------------------------------------------------------
// Kernel 2b: mB[b] = max_j s_j[b, j]
// ---------------------------------------------------------------------------
__global__ __launch_bounds__(256) void bmax_kernel(const float* __restrict__ s_j,
                                                   float* __restrict__ mB) {
    __shared__ float red[256];
    const int b = blockIdx.x;
    float m = -1e30f;
    for (int n = threadIdx.x; n < N_; n += 256)
        m = fmaxf(m, s_j[b * N_ + n]);
    red[threadIdx.x] = m;
    __syncthreads();
    for (int s = 128; s > 0; s >>= 1) {
        if (threadIdx.x < s)
            red[threadIdx.x] = fmaxf(red[threadIdx.x], red[threadIdx.x + s]);
        __syncthreads();
    }
    if (threadIdx.x == 0) mB[b] = red[0];
}

// ---------------------------------------------------------------------------
// Kernel 3: flash-style aggregation with exact precomputed row max
//   (lrelu monotone => m_i = lrelu(s_i + max_j s_j)).
// 4 waves share each async-staged, double-buffered j-tile in LDS;
// wave w owns output col-tiles {2w, 2w+1}. 8 WMMAs / wave / j-tile.
// ---------------------------------------------------------------------------
__global__ __launch_bounds__(128) void attn_kernel(const float* __restrict__ Wh,
                                                   const float* __restrict__ s_i,
                                                   const float* __restrict__ s_j,
                                                   const float* __restrict__ mB,
                                                   float* __restrict__ out) {
    __shared__ float sjs[N_];               // s_j[b,:]  (8 KB)
    __shared__ float tile[2][16 * TSTR];    // double-buffered j-tile (2 x 8.25 KB)

    const int t    = threadIdx.x;
    const int b    = blockIdx.x >> 7;       // N_/16 = 128 i-tiles per batch
    const int i0   = (blockIdx.x & 127) * 16;
    const int wave = t >> 5;
    const int lane = t & 31;
    const int half = lane >> 4;
    const int lm   = lane & 15;

    const float si_row = s_i[b * N_ + i0 + lm];   // P-row for this lane (A layout M = lane&15)
    const float m      = lrelu(si_row + mB[b]);   // exact row max
    const float* whb   = Wh + (size_t)b * N_ * FOUT_;

    // stage s_j[b,:] into LDS (plain vector copies; visible after first barrier)
#pragma unroll
    for (int r = 0; r < 4; ++r) {
        int e = r * 512 + t * 4;
        *(float4*)(sjs + e) = *(const float4*)(s_j + b * N_ + e);
    }

    // kick off async stage of j-tile 0 (4 x b128 per wave)
#pragma unroll
    for (int r = 0; r < 4; ++r) {
        int e = r * 512 + t * 4;
        async_b128(whb + (e >> 7) * FOUT_ + (e & 127),
                   tile[0] + (e >> 7) * TSTR + (e & 127));
    }

    float l = 0.f;
    v8f acc[2] = {};

    for (int jt = 0; jt < NT_; ++jt) {
        if (jt + 1 < NT_) {
            __syncthreads();                 // all reads of tile[(jt+1)&1] (iter jt-1) done
            const float* wn = whb + (size_t)(jt + 1) * 16 * FOUT_;
            float* bn = tile[(jt + 1) & 1];
#pragma unroll
            for (int r = 0; r < 4; ++r) {
                int e = r * 512 + t * 4;
                async_b128(wn + (e >> 7) * FOUT_ + (e & 127),
                           bn + (e >> 7) * TSTR + (e & 127));
            }
            wait_async<4>();                 // per-wave FIFO: tile jt's 4 ops retired
        } else {
            wait_async<0>();
        }
        __syncthreads();                     // all waves' portions of tile jt landed

        const int j0 = jt * 16;
        const float* buf = tile[jt & 1];

        // P-tile as four 16x4 A-operands; lane covers cols 4q + 2*half + {0,1}
        v2f p[4];
#pragma unroll
        for (int q = 0; q < 4; ++q) {
            const int c0 = 4 * q + 2 * half;
            float p0 = __expf(lrelu(si_row + sjs[j0 + c0])     - m);
            float p1 = __expf(lrelu(si_row + sjs[j0 + c0 + 1]) - m);
            p[q].x = p0;
            p[q].y = p1;
            l += p0 + p1;
        }
        // acc[16 x 32] += P[16x16] @ tile[16 x 32(cols of this wave)]
#pragma unroll
        for (int ct = 0; ct < 2; ++ct) {
            const int c = 2 * wave + ct;
#pragma unroll
            for (int q = 0; q < 4; ++q) {
                const int kr = 4 * q + 2 * half;
                v2f bb;
                bb.x = buf[kr * TSTR + c * 16 + lm];
                bb.y = buf[(kr + 1) * TSTR + c * 16 + lm];   // fuses to ds_load_2addr
                acc[ct] = __builtin_amdgcn_wmma_f32_16x16x4_f32(false, p[q], false, bb,
                                                                (short)0, acc[ct], false, false);
            }
        }
    }

    // full row sums live per wave: combine lane halves, broadcast per output row
    l += __shfl_xor(l, 16, 32);
#pragma unroll
    for (int ct = 0; ct < 2; ++ct) {
        const int c = 2 * wave + ct;
#pragma unroll
        for (int r = 0; r < 8; ++r) {
            const int row = r + 8 * half;
            const float lr = __shfl(l, row, 32);
            out[((size_t)b * N_ + i0 + row) * FOUT_ + c * 16 + lm] = acc[ct][r] / lr;
        }
    }
}

// ---------------------------------------------------------------------------
extern "C" void kernel_launch(void* const* d_in, const int* in_sizes, int n_in,
                              void* d_out, int out_size, void* d_ws, size_t ws_size,
                              hipStream_t stream) {
    const float* h   = (const float*)d_in[0];   // [B, N, FIN]
    const float* W   = (const float*)d_in[1];   // [FIN, FOUT]
    const float* a_i = (const float*)d_in[2];   // [FOUT]
    const float* a_j = (const float*)d_in[3];   // [FOUT]
    float* out = (float*)d_out;                 // [B, N, FOUT]

    float* Wh  = (float*)d_ws;                          // 8 MB
    float* s_i = Wh + (size_t)B_ * N_ * FOUT_;          // 64 KB
    float* s_j = s_i + (size_t)B_ * N_;                 // 64 KB
    float* mB  = s_j + (size_t)B_ * N_;                 // 32 B

    wh_kernel   <<<B_ * N_ / 16, 256, 0, stream>>>(h, W, Wh);
    score_kernel<<<B_ * N_ / 8,  256, 0, stream>>>(Wh, a_i, a_j, s_i, s_j);
    bmax_kernel <<<B_,           256, 0, stream>>>(s_j, mB);
    attn_kernel <<<B_ * N_ / 16, 128, 0, stream>>>(Wh, s_i, s_j, mB, out);
}